// GeometryAwareAttention_6897717477665
// MI455X (gfx1250) — compile-verified
//
#include <hip/hip_runtime.h>
#include <hip/hip_bf16.h>
#include <math.h>

// ---------------------------------------------------------------------------
// GeometryAwareAttention for MI455X (gfx1250, wave32, WMMA).
//
// B=2, N=512, DIM=256, HEADS=4, D_HEAD=64, D_G=64, SCALE=1/8.
//
// Roofline: geom (134 MB) read once + ~25 MB scratch traffic -> ~7 us floor at
// 23.3 TB/s. Total ~1.6 GFLOP; on V_WMMA_F32_16X16X4_F32 (~2048 FLOP/instr)
// compute is ~1.5 us device-wide, fully hidden under memory. So: keep f32
// precision end-to-end, fuse geom algebraically (never materialize geom@Wkg),
// and put every contraction on the WMMA pipe.
//
// Fusion algebra:
//   q . geom_k[b,i,j,h] = sum_g geom[b,i,j,g] * wq[b,i,h,g],
//     wq[b,i,h,g] = sum_d Q[b,i,h,d] * Wkg[g, h*64+d]
//   wcomb[b,i,h,g] = SCALE*wq + Wgb[g,h]   (folds the Wgb logit term)
//   bkg term and log(c_i) term are constant along softmax axis j -> cancel.
// ---------------------------------------------------------------------------

#define BB    2
#define NN    512
#define DIM   256
#define HEADS 4
#define DH    64
#define DG    64
#define SCALE_F 0.125f

typedef float v2f __attribute__((ext_vector_type(2)));
typedef float v8f __attribute__((ext_vector_type(8)));

// D = A(16x4 f32) * B(4x16 f32) + C(16x16 f32), full-rate f32 matrix pipe.
#define WMMA4(a, b, c) \
  __builtin_amdgcn_wmma_f32_16x16x4_f32(false, (a), false, (b), (short)0, (c), false, false)

// Per the CDNA5 ISA A-matrix 16x4 layout: lanes 0-15 hold K={0,1} in the two
// A VGPRs, lanes 16-31 hold K={2,3}. We mirror that for B (rows striped across
// lanes within a VGPR). With ml = lane&15, kh = lane>>4:
//   a = (A[ml][k+2*kh], A[ml][k+2*kh+1])   -- contiguous along K
//   b = (B[k+2*kh][n0+ml], B[k+2*kh+1][n0+ml])
// C/D: vgpr r, lane l -> row (r + 8*(l>=16)), col (l&15).

// ---------------------------------------------------------------------------
// Kernel 1: fused Q/K/V projection. C(1024x256) = X @ W + b, three weight sets.
// One wave per 16x16 output tile. 64 WMMA steps over K=256.
// ---------------------------------------------------------------------------
__global__ void qkv_gemm(const float* __restrict__ x,
                         const float* __restrict__ Wq, const float* __restrict__ bq,
                         const float* __restrict__ Wk, const float* __restrict__ bk,
                         const float* __restrict__ Wv, const float* __restrict__ bv,
                         float* __restrict__ Q, float* __restrict__ K,
                         float* __restrict__ V) {
  const int wave = (blockIdx.x * blockDim.x + threadIdx.x) >> 5;
  const int lane = threadIdx.x & 31;
  int t = wave;
  const int ntile = t & 15;  t >>= 4;        // 16 N-tiles (256/16)
  const int mtile = t & 63;  t >>= 6;        // 64 M-tiles (1024/16)
  const int mat   = t;                       // 0=Q 1=K 2=V
  const float* W    = (mat == 0) ? Wq : (mat == 1) ? Wk : Wv;
  const float* bias = (mat == 0) ? bq : (mat == 1) ? bk : bv;
  float*       Out  = (mat == 0) ? Q  : (mat == 1) ? K  : V;

  const int m0 = mtile * 16, n0 = ntile * 16;
  const int ml = lane & 15, kh = lane >> 4;

  v8f c = {0.f, 0.f, 0.f, 0.f, 0.f, 0.f, 0.f, 0.f};
  const float* arow = x + (m0 + ml) * DIM;
  for (int k = 0; k < DIM; k += 4) {
    v2f a = *(const v2f*)(arow + k + 2 * kh);
    v2f b;
    b.x = W[(k + 2 * kh)     * DIM + n0 + ml];
    b.y = W[(k + 2 * kh + 1) * DIM + n0 + ml];
    c = WMMA4(a, b, c);
  }
  const float bn = bias[n0 + ml];
#pragma unroll
  for (int r = 0; r < 8; ++r)
    Out[(m0 + r + 8 * kh) * DIM + n0 + ml] = c[r] + bn;
}

// ---------------------------------------------------------------------------
// Kernel 2: wcomb[bi,h,g] = SCALE * sum_d Q[bi,h*64+d]*Wkg[g,h*64+d] + Wgb[g,h]
// One block per (b,i); 256 threads = (h,g). 33 MFLOP total -> VALU is fine.
// ---------------------------------------------------------------------------
__global__ void prep_wcomb(const float* __restrict__ Q,
                           const float* __restrict__ Wkg,
                           const float* __restrict__ Wgb,
                           float* __restrict__ wcomb) {
  __shared__ float qrow[DIM];
  const int bi = blockIdx.x;
  const int t  = threadIdx.x;
  qrow[t] = Q[bi * DIM + t];
  __syncthreads();
  const int h = t >> 6, g = t & 63;
  const float* wr = Wkg + g * DIM + h * DH;   // Wkg is (DG, DIM) row-major
  const float* qr = qrow + h * DH;
  float acc = 0.f;
#pragma unroll 8
  for (int d = 0; d < DH; ++d) acc += qr[d] * wr[d];
  wcomb[(bi * HEADS + h) * DG + g] = acc * SCALE_F + Wgb[g * HEADS + h];
}

// logc[b*N+j] = gamma * log(conf + eps); the log(c_i) half cancels in softmax.
__global__ void prep_logc(const float* __restrict__ conf,
                          const float* __restrict__ gamma,
                          float* __restrict__ logc) {
  const int i = blockIdx.x * blockDim.x + threadIdx.x;
  if (i < BB * NN) logc[i] = gamma[0] * __logf(conf[i] + 1e-6f);
}

// ---------------------------------------------------------------------------
// Kernel 3: S[b,h,i,j] = SCALE * sum_d Q[b,i,h,d]*K[b,j,h,d].  Per (b,h):
// 512x512x64 GEMM. One wave per 16x16 tile, 16 WMMA steps; both operands are
// contiguous float2 loads along d.
// ---------------------------------------------------------------------------
__global__ void qk_gemm(const float* __restrict__ Q, const float* __restrict__ K,
                        float* __restrict__ S) {
  const int wave = (blockIdx.x * blockDim.x + threadIdx.x) >> 5;
  const int lane = threadIdx.x & 31;
  int t = wave;
  const int jt = t & 31; t >>= 5;
  const int it = t & 31; t >>= 5;
  const int bh = t;                    // 0..7
  const int b = bh >> 2, h = bh & 3;
  const int i0 = it * 16, j0 = jt * 16;
  const int ml = lane & 15, kh = lane >> 4;

  const float* Qr = Q + (b * NN + i0 + ml) * DIM + h * DH;
  const float* Kr = K + (b * NN + j0 + ml) * DIM + h * DH;   // lane ml = col n
  v8f c = {0.f, 0.f, 0.f, 0.f, 0.f, 0.f, 0.f, 0.f};
  for (int d = 0; d < DH; d += 4) {
    v2f a  = *(const v2f*)(Qr + d + 2 * kh);
    v2f bb = *(const v2f*)(Kr + d + 2 * kh);
    c = WMMA4(a, bb, c);
  }
  float* Sp = S + ((size_t)bh * NN + i0) * NN + j0;
#pragma unroll
  for (int r = 0; r < 8; ++r)
    Sp[(r + 8 * kh) * NN + ml] = c[r] * SCALE_F;
}

// ---------------------------------------------------------------------------
// Kernel 4: per (b,i) block: geom logit term via WMMA (A = wcomb in padded LDS,
// B = 16 contiguous geom rows), add qk logits + confidence bias, softmax over
// j, write attention in-place over S. This kernel carries the 134 MB geom read.
// ---------------------------------------------------------------------------
#define LGS 516   // 512 + pad
__global__ void geom_softmax(const float* __restrict__ geom,
                             const float* __restrict__ wcomb,
                             const float* __restrict__ logc,
                             float* __restrict__ S /* in: qk logits, out: attn */) {
  __shared__ __align__(16) float wc[16 * 68];   // 16 rows (heads 0-3, rest 0), pad 68
  __shared__ float lg[HEADS * LGS];
  __shared__ float red[HEADS * 64];
  const int bi = blockIdx.x;                    // b*N + i
  const int b  = bi >> 9, i = bi & 511;
  const int t  = threadIdx.x;                   // 256 threads = 8 waves
  const int wave = t >> 5, lane = t & 31;
  const int ml = lane & 15, kh = lane >> 4;

  // Stage wcomb rows as the A matrix (M=16 with rows >=4 zeroed).
  for (int idx = t; idx < 16 * 68; idx += 256) {
    const int row = idx / 68, col = idx - row * 68;
    wc[idx] = (row < HEADS && col < DG) ? wcomb[(bi * HEADS + row) * DG + col] : 0.f;
  }
  __syncthreads();

  // Phase A: logits. Each wave handles 4 j-tiles of 16; geom rows for fixed
  // (b,i) are a contiguous 128 KB stream -> read exactly once, prefetch ahead.
  const float* grow = geom + (size_t)bi * NN * DG;
  const float* aA = wc + ml * 68;
  for (int q = 0; q < 4; ++q) {
    const int j0 = (wave * 4 + q) * 16;
    const float* gB = grow + (size_t)(j0 + ml) * DG;
    __builtin_prefetch(gB + 16 * DG, 0, 0);     // global_prefetch for next tile
    v8f c = {0.f, 0.f, 0.f, 0.f, 0.f, 0.f, 0.f, 0.f};
    for (int g = 0; g < DG; g += 4) {
      v2f a  = *(const v2f*)(aA + g + 2 * kh);  // LDS, conflict-free via pad
      v2f bb = *(const v2f*)(gB + g + 2 * kh);  // contiguous geom float2
      c = WMMA4(a, bb, c);
    }
    if (kh == 0) {                               // rows 0-3 = heads
#pragma unroll
      for (int r = 0; r < HEADS; ++r) {
        const int j = j0 + ml;
        const float qk = S[(((size_t)b * HEADS + r) * NN + i) * NN + j];
        lg[r * LGS + j] = c[r] + qk + logc[b * NN + j];
      }
    }
  }
  __syncthreads();

  // Phase B: softmax over j per head. 64 threads per head, LDS tree-reduce.
  const int h = t >> 6, l = t & 63;
  float m = -INFINITY;
  for (int j = l; j < NN; j += 64) m = fmaxf(m, lg[h * LGS + j]);
  red[h * 64 + l] = m;
  __syncthreads();
  for (int s = 32; s > 0; s >>= 1) {
    if (l < s) red[h * 64 + l] = fmaxf(red[h * 64 + l], red[h * 64 + l + s]);
    __syncthreads();
  }
  const float rm = red[h * 64];
  __syncthreads();
  float sum = 0.f;
  for (int j = l; j < NN; j += 64) {
    const float e = __expf(lg[h * LGS + j] - rm);
    lg[h * LGS + j] = e;
    sum += e;
  }
  red[h * 64 + l] = sum;
  __syncthreads();
  for (int s = 32; s > 0; s >>= 1) {
    if (l < s) red[h * 64 + l] += red[h * 64 + l + s];
    __syncthreads();
  }
  const float inv = 1.f / red[h * 64];
  for (int j = l; j < NN; j += 64)
    S[(((size_t)b * HEADS + h) * NN + i) * NN + j] = lg[h * LGS + j] * inv;
}

// ---------------------------------------------------------------------------
// Kernel 5: O[b,i,h*64+d] = sum_j P[b,h,i,j] * V[b,j,h*64+d].  Per (b,h):
// 512x64x512 GEMM; one wave per 16x16 tile, 128 WMMA steps.
// ---------------------------------------------------------------------------
__global__ void av_gemm(const float* __restrict__ P, const float* __restrict__ V,
                        float* __restrict__ O) {
  const int wave = (blockIdx.x * blockDim.x + threadIdx.x) >> 5;
  const int lane = threadIdx.x & 31;
  int t = wave;
  const int dt = t & 3;  t >>= 2;      // 4 tiles over d (64/16)
  const int it = t & 31; t >>= 5;      // 32 tiles over i
  const int bh = t;
  const int b = bh >> 2, h = bh & 3;
  const int i0 = it * 16, d0 = dt * 16;
  const int ml = lane & 15, kh = lane >> 4;

  const float* Pr = P + ((size_t)bh * NN + i0 + ml) * NN;
  v8f c = {0.f, 0.f, 0.f, 0.f, 0.f, 0.f, 0.f, 0.f};
  for (int j = 0; j < NN; j += 4) {
    v2f a = *(const v2f*)(Pr + j + 2 * kh);
    v2f bb;
    bb.x = V[((size_t)b * NN + j + 2 * kh)     * DIM + h * DH + d0 + ml];
    bb.y = V[((size_t)b * NN + j + 2 * kh + 1) * DIM + h * DH + d0 + ml];
    c = WMMA4(a, bb, c);
  }
#pragma unroll
  for (int r = 0; r < 8; ++r)
    O[((size_t)b * NN + i0 + r + 8 * kh) * DIM + h * DH + d0 + ml] = c[r];
}

// ---------------------------------------------------------------------------
// Kernel 6: out = O @ Wo + bo. 1024x256x256 GEMM.
// ---------------------------------------------------------------------------
__global__ void out_gemm(const float* __restrict__ O, const float* __restrict__ Wo,
                         const float* __restrict__ bo, float* __restrict__ out) {
  const int wave = (blockIdx.x * blockDim.x + threadIdx.x) >> 5;
  const int lane = threadIdx.x & 31;
  int t = wave;
  const int ntile = t & 15; t >>= 4;
  const int mtile = t;                 // 0..63
  const int m0 = mtile * 16, n0 = ntile * 16;
  const int ml = lane & 15, kh = lane >> 4;

  v8f c = {0.f, 0.f, 0.f, 0.f, 0.f, 0.f, 0.f, 0.f};
  const float* arow = O + (m0 + ml) * DIM;
  for (int k = 0; k < DIM; k += 4) {
    v2f a = *(const v2f*)(arow + k + 2 * kh);
    v2f b;
    b.x = Wo[(k + 2 * kh)     * DIM + n0 + ml];
    b.y = Wo[(k + 2 * kh + 1) * DIM + n0 + ml];
    c = WMMA4(a, b, c);
  }
  const float bn = bo[n0 + ml];
#pragma unroll
  for (int r = 0; r < 8; ++r)
    out[(m0 + r + 8 * kh) * DIM + n0 + ml] = c[r] + bn;
}

// ---------------------------------------------------------------------------
extern "C" void kernel_launch(void* const* d_in, const int* in_sizes, int n_in,
                              void* d_out, int out_size, void* d_ws, size_t ws_size,
                              hipStream_t stream) {
  const float* x     = (const float*)d_in[0];
  const float* geom  = (const float*)d_in[1];
  const float* conf  = (const float*)d_in[2];
  const float* Wq    = (const float*)d_in[3];
  const float* bq    = (const float*)d_in[4];
  const float* Wk    = (const float*)d_in[5];
  const float* bk    = (const float*)d_in[6];
  const float* Wv    = (const float*)d_in[7];
  const float* bv    = (const float*)d_in[8];
  const float* Wkg   = (const float*)d_in[9];
  // d_in[10] = bkg: constant along softmax axis -> cancels, unused.
  const float* Wgb   = (const float*)d_in[11];
  const float* gamma = (const float*)d_in[12];
  const float* Wo    = (const float*)d_in[13];
  const float* bo    = (const float*)d_in[14];
  float* out = (float*)d_out;

  // Workspace layout (floats): Q,K,V,O (1024x256 each), wcomb (1024x4x64),
  // logc (1024), S (2x4x512x512, reused in-place for attention). ~13.6 MB.
  float* ws    = (float*)d_ws;
  float* Q     = ws;
  float* K     = Q + 1024 * 256;
  float* V     = K + 1024 * 256;
  float* O     = V + 1024 * 256;
  float* wcomb = O + 1024 * 256;
  float* logc  = wcomb + 1024 * 256;
  float* S     = logc + 1024;

  qkv_gemm    <<<768,  128, 0, stream>>>(x, Wq, bq, Wk, bk, Wv, bv, Q, K, V);
  prep_wcomb  <<<1024, 256, 0, stream>>>(Q, Wkg, Wgb, wcomb);
  prep_logc   <<<4,    256, 0, stream>>>(conf, gamma, logc);
  qk_gemm     <<<2048, 128, 0, stream>>>(Q, K, S);
  geom_softmax<<<1024, 256, 0, stream>>>(geom, wcomb, logc, S);
  av_gemm     <<<256,  128, 0, stream>>>(S, V, O);
  out_gemm    <<<256,  128, 0, stream>>>(O, Wo, bo, out);
}